// TensorNeRF_45457933861365
// MI455X (gfx1250) — compile-verified
//
#include <hip/hip_runtime.h>
#include <hip/hip_bf16.h>

typedef __attribute__((ext_vector_type(2))) float v2f;
typedef __attribute__((ext_vector_type(8))) float v8f;

#define N_RAYS 4096
#define N_SAMP 512
#define TOPK   4000u

constexpr float STEP_SIZE_F = 0.010148735200599281f;   // 3*sqrt(3)/512
constexpr float DIST_F      = 0.25371838001498203f;    // STEP*25
constexpr float ALPHA_THRES = 0.001f;
constexpr float W_THRES     = 0.0001f;

// d_out layout (floats): weights | bg_weight | depth | full_mask | inv_mask
#define W_OFF     0
#define BG_OFF    (N_RAYS * N_SAMP)            // 2097152
#define DEPTH_OFF (BG_OFF + N_RAYS)            // 2101248
#define FULL_OFF  (DEPTH_OFF + N_RAYS)         // 2105344
#define INV_OFF   (FULL_OFF + N_RAYS * N_SAMP) // 4202496
#define N_SCORE   (N_RAYS * N_SAMP)

#if defined(__gfx1250__) && __has_builtin(__builtin_amdgcn_wmma_f32_16x16x4_f32)
#define USE_WMMA 1
#else
#define USE_WMMA 0
#endif

// ---------------- trilinear gather helpers ----------------
struct TL { int i; float xd, yd, zd; };

__device__ inline TL make_tl(int R, float nx, float ny, float nz) {
    float fx = (fminf(fmaxf(nx, -1.f), 1.f) + 1.f) * 0.5f * (float)(R - 1);
    float fy = (fminf(fmaxf(ny, -1.f), 1.f) + 1.f) * 0.5f * (float)(R - 1);
    float fz = (fminf(fmaxf(nz, -1.f), 1.f) + 1.f) * 0.5f * (float)(R - 1);
    int x0 = (int)floorf(fx); x0 = x0 < 0 ? 0 : (x0 > R - 2 ? R - 2 : x0);
    int y0 = (int)floorf(fy); y0 = y0 < 0 ? 0 : (y0 > R - 2 ? R - 2 : y0);
    int z0 = (int)floorf(fz); z0 = z0 < 0 ? 0 : (z0 > R - 2 ? R - 2 : z0);
    TL t;
    t.xd = fx - (float)x0; t.yd = fy - (float)y0; t.zd = fz - (float)z0;
    t.i = (z0 * R + y0) * R + x0;
    return t;
}

__device__ inline float gather8(const float* __restrict__ v, const TL& t, int R) {
    int sy = R, sz = R * R, i = t.i;
    float c000 = v[i],          c001 = v[i + 1];
    float c010 = v[i + sy],     c011 = v[i + sy + 1];
    float c100 = v[i + sz],     c101 = v[i + sz + 1];
    float c110 = v[i + sz + sy],c111 = v[i + sz + sy + 1];
    float c00 = c000 * (1.f - t.xd) + c001 * t.xd;
    float c01 = c010 * (1.f - t.xd) + c011 * t.xd;
    float c10 = c100 * (1.f - t.xd) + c101 * t.xd;
    float c11 = c110 * (1.f - t.xd) + c111 * t.xd;
    float c0 = c00 * (1.f - t.yd) + c01 * t.yd;
    float c1 = c10 * (1.f - t.yd) + c11 * t.yd;
    return c0 * (1.f - t.zd) + c1 * t.zd;
}

// ---------------- main ray-march kernel: 1 wave per ray ----------------
__global__ __launch_bounds__(256) void k_march(
    const float* __restrict__ ro, const float* __restrict__ rd,
    const float* __restrict__ dens, const float* __restrict__ prob,
    const float* __restrict__ alvol, float* __restrict__ out)
{
    __shared__ float sZ[8 * N_SAMP];   // log(1-alpha+eps), then prefix sums P
    __shared__ float sA[8 * N_SAMP];   // per-sample alpha
    __shared__ float sOff[8 * 32];     // per-column scan offsets

    const int lane = threadIdx.x & 31;
    const int wave = threadIdx.x >> 5;
    const int ray  = blockIdx.x * 8 + wave;
    float* zw = &sZ[wave * N_SAMP];
    float* za = &sA[wave * N_SAMP];

    // ray/bbox intersection (uniform per wave; redundant per lane is cheap)
    float ox = ro[ray * 3 + 0], oy = ro[ray * 3 + 1], oz = ro[ray * 3 + 2];
    float dx = rd[ray * 3 + 0], dy = rd[ray * 3 + 1], dz = rd[ray * 3 + 2];
    float vx = (dx == 0.f) ? 1e-6f : dx;
    float vy = (dy == 0.f) ? 1e-6f : dy;
    float vz = (dz == 0.f) ? 1e-6f : dz;
    float tmin = fminf((1.5f - ox) / vx, (-1.5f - ox) / vx);
    tmin = fmaxf(tmin, fminf((1.5f - oy) / vy, (-1.5f - oy) / vy));
    tmin = fmaxf(tmin, fminf((1.5f - oz) / vz, (-1.5f - oz) / vz));
    tmin = fminf(fmaxf(tmin, 2.0f), 6.0f);

    // Pass A: alpha-volume + density gathers; lane L covers s = j*32+L so the
    // 32 lanes of each iteration sample 32 consecutive ray positions (L2-friendly)
    for (int j = 0; j < 16; ++j) {
        int s = j * 32 + lane;
        float t = tmin + STEP_SIZE_F * (float)s;
        float px = ox + dx * t, py = oy + dy * t, pz = oz + dz * t;
        bool inb = (px >= -1.5f) & (px <= 1.5f) & (py >= -1.5f) & (py <= 1.5f)
                 & (pz >= -1.5f) & (pz <= 1.5f);
        float al = 0.f;
        if (inb) {
            float nx = px * (1.f / 1.5f), ny = py * (1.f / 1.5f), nz = pz * (1.f / 1.5f);
            TL t256 = make_tl(256, nx, ny, nz);
            __builtin_prefetch(dens + t256.i, 0, 1);        // global_prefetch_b8
            float a = gather8(alvol, make_tl(128, nx, ny, nz), 128);
            if (a > ALPHA_THRES) {
                float x = gather8(dens, t256, 256) - 10.0f; // feat + DENSITY_SHIFT
                float sig = (x > 20.f) ? x : log1pf(expf(x));
                al = 1.f - expf(-sig * DIST_F);
            }
        }
        zw[s] = log1pf(1e-10f - al);   // log(1 - alpha + 1e-10), exact near 0
        za[s] = al;
    }
    __syncthreads();

    // Scan: cumprod in log space.  View zw as Z[16 rows][32 cols], col c =
    // samples c*16..c*16+15.  P = L16 x Z via V_WMMA_F32_16X16X4_F32.
#if USE_WMMA
    {
        const int m = lane & 15;
        const int khalf = (lane < 16) ? 0 : 2;
        v8f acc0 = {0.f,0.f,0.f,0.f,0.f,0.f,0.f,0.f};
        v8f acc1 = {0.f,0.f,0.f,0.f,0.f,0.f,0.f,0.f};
#pragma unroll
        for (int kk = 0; kk < 4; ++kk) {
            int kb = 4 * kk + khalf;
            v2f av;  // A = lower-triangular ones, M in lanes, K in regs
            av.x = ((float)(kb     <= m)); av.y = ((float)(kb + 1 <= m));
            v2f b0, b1;  // B = Z chunk, N in lanes, K split like A
            b0.x = zw[m * 16 + kb];          b0.y = zw[m * 16 + kb + 1];
            b1.x = zw[(m + 16) * 16 + kb];   b1.y = zw[(m + 16) * 16 + kb + 1];
            acc0 = __builtin_amdgcn_wmma_f32_16x16x4_f32(false, av, false, b0,
                                                         (short)0, acc0, false, false);
            acc1 = __builtin_amdgcn_wmma_f32_16x16x4_f32(false, av, false, b1,
                                                         (short)0, acc1, false, false);
        }
        __syncthreads();  // all B reads done before overwriting zw with P
        int rbase = (lane < 16) ? 0 : 8;
#pragma unroll
        for (int v = 0; v < 8; ++v) {
            zw[m * 16 + (v + rbase)]        = acc0[v];
            zw[(m + 16) * 16 + (v + rbase)] = acc1[v];
        }
    }
#else
    {   // fallback: lane c scans its 16-sample column serially
        float run = 0.f;
        for (int r = 0; r < 16; ++r) { run += zw[lane * 16 + r]; zw[lane * 16 + r] = run; }
    }
#endif
    __syncthreads();

    // Exclusive prefix of the 32 column totals (P row 15) via wave32 shuffle
    // scan: one coalesced ds_load per lane + 5 shfl steps (no divergent DS loop).
    {
        float tot  = zw[lane * 16 + 15];
        float incl = tot;
#pragma unroll
        for (int d = 1; d < 32; d <<= 1) {
            float n = __shfl_up(incl, d, 32);
            if (lane >= d) incl += n;
        }
        sOff[wave * 32 + lane] = incl - tot;     // exclusive offset
        if (lane == 31)                          // bg_weight = exp(total log-sum)
            out[BG_OFF + ray] = expf(incl);
    }
    __syncthreads();

    // Pass B: weights, depth, scores (prob gathered only where weight > thres)
    float depth = 0.f;
    for (int j = 0; j < 16; ++j) {
        int s = j * 32 + lane;
        int c = s >> 4, r = s & 15;
        float cumE = ((r > 0) ? zw[c * 16 + r - 1] : 0.f) + sOff[wave * 32 + c];
        float w = za[s] * expf(cumE);
        float t = tmin + STEP_SIZE_F * (float)s;
        depth += w * t;
        float score = 0.f;
        if (w > W_THRES) {
            float px = ox + dx * t, py = oy + dy * t, pz = oz + dz * t;
            TL t256 = make_tl(256, px * (1.f/1.5f), py * (1.f/1.5f), pz * (1.f/1.5f));
            score = w * gather8(prob, t256, 256);
        }
        out[W_OFF + ray * N_SAMP + s]    = w;
        out[FULL_OFF + ray * N_SAMP + s] = score;  // staged score, masked later
    }
#pragma unroll
    for (int m = 16; m >= 1; m >>= 1) depth += __shfl_xor(depth, m, 32);
    if (lane == 0) out[DEPTH_OFF + ray] = depth;
}

// ---------------- top-k = 3-pass radix select on positive float bits ----------------
__global__ void k_init(unsigned int* hist, unsigned int* state) {
    int i = blockIdx.x * blockDim.x + threadIdx.x;
    if (i < 3 * 2048) hist[i] = 0u;
    if (i < 8) state[i] = 0u;
}

__global__ __launch_bounds__(256) void k_hist(
    const float* __restrict__ scores, unsigned int* __restrict__ hist,
    const unsigned int* __restrict__ state, int pass)
{
    __shared__ unsigned int lh[2048];
    for (int i = threadIdx.x; i < 2048; i += 256) lh[i] = 0u;
    __syncthreads();
    unsigned int b1 = state[0], b2 = state[2];
    int stride = gridDim.x * 256;
    for (int i = blockIdx.x * 256 + threadIdx.x; i < N_SCORE; i += stride) {
        unsigned int bits = __float_as_uint(scores[i]);  // scores >= 0
        if (pass == 0) {
            atomicAdd(&lh[bits >> 21], 1u);
        } else if (pass == 1) {
            if ((bits >> 21) == b1) atomicAdd(&lh[(bits >> 10) & 0x7FFu], 1u);
        } else {
            if ((bits >> 21) == b1 && ((bits >> 10) & 0x7FFu) == b2)
                atomicAdd(&lh[bits & 0x3FFu], 1u);
        }
    }
    __syncthreads();
    for (int i = threadIdx.x; i < 2048; i += 256)
        if (lh[i]) atomicAdd(&hist[pass * 2048 + i], lh[i]);
}

__global__ void k_pick(const unsigned int* __restrict__ hist,
                       unsigned int* __restrict__ state, int pass)
{
    if (threadIdx.x | blockIdx.x) return;
    unsigned int K = (pass == 0) ? TOPK : state[pass * 2 - 1];
    int nb = (pass == 2) ? 1024 : 2048;
    unsigned int base = pass * 2048, cum = 0, chosen = 0, krem = 0;
    for (int b = nb - 1; b >= 0; --b) {
        unsigned int c = hist[base + b];
        if (cum + c >= K) { chosen = (unsigned int)b; krem = K - cum; break; }
        cum += c;
    }
    state[pass * 2] = chosen;
    state[pass * 2 + 1] = krem;
    if (pass == 2) state[6] = (state[0] << 21) | (state[2] << 10) | chosen;
}

__global__ __launch_bounds__(256) void k_mask(float* __restrict__ out,
                                              const unsigned int* __restrict__ state)
{
    int i = blockIdx.x * 256 + threadIdx.x;
    if (i >= N_SCORE) return;
    float thr = __uint_as_float(state[6]);   // value of the 4000th-largest score
    float w  = out[W_OFF + i];
    float sc = out[FULL_OFF + i];
    bool full = (sc > 0.f) && (sc >= thr);
    out[FULL_OFF + i] = full ? 1.f : 0.f;
    out[INV_OFF + i]  = ((w > W_THRES) && !full) ? 1.f : 0.f;
}

// ---------------- launcher ----------------
extern "C" void kernel_launch(void* const* d_in, const int* in_sizes, int n_in,
                              void* d_out, int out_size, void* d_ws, size_t ws_size,
                              hipStream_t stream)
{
    const float* ro    = (const float*)d_in[0];
    const float* rd    = (const float*)d_in[1];
    const float* dens  = (const float*)d_in[2];
    const float* prob  = (const float*)d_in[3];
    const float* alvol = (const float*)d_in[4];
    float* out = (float*)d_out;

    unsigned int* hist  = (unsigned int*)d_ws;          // 3*2048 u32
    unsigned int* state = hist + 3 * 2048;              // 8 u32

    k_init<<<dim3(24), dim3(256), 0, stream>>>(hist, state);
    k_march<<<dim3(N_RAYS / 8), dim3(256), 0, stream>>>(ro, rd, dens, prob, alvol, out);

    const float* scores = out + FULL_OFF;
    k_hist<<<dim3(1024), dim3(256), 0, stream>>>(scores, hist, state, 0);
    k_pick<<<dim3(1), dim3(1), 0, stream>>>(hist, state, 0);
    k_hist<<<dim3(1024), dim3(256), 0, stream>>>(scores, hist, state, 1);
    k_pick<<<dim3(1), dim3(1), 0, stream>>>(hist, state, 1);
    k_hist<<<dim3(1024), dim3(256), 0, stream>>>(scores, hist, state, 2);
    k_pick<<<dim3(1), dim3(1), 0, stream>>>(hist, state, 2);
    k_mask<<<dim3(N_SCORE / 256), dim3(256), 0, stream>>>(out, state);
}